// GraphConvolution_2027224564235
// MI455X (gfx1250) — compile-verified
//
#include <hip/hip_runtime.h>

// ---------------------------------------------------------------------------
// GCN layer: out = segment_sum(val * (x@W)[col] -> row) + b
//   x: [N,256] f32, W: [256,128] f32, edges: int64 row/col + f32 val
// Phase 1 (WMMA GEMM): support = x @ W  (v_wmma_f32_16x16x4_f32), d_ws scratch
// Phase 2 (SpMM):      wave-per-edge gather + f32 global atomic scatter
// ---------------------------------------------------------------------------

typedef float v2f __attribute__((ext_vector_type(2)));
typedef float v8f __attribute__((ext_vector_type(8)));

#define IN_F   256
#define OUT_F  128
#define KC      32   // K-chunk staged in LDS per iteration
#define MT      32   // rows (M) per block: two 16-row WMMA tiles

// ---------------------------------------------------------------- init + bias
__global__ __launch_bounds__(256) void gcn_init_out(const float* __restrict__ b,
                                                    float* __restrict__ out,
                                                    int total) {
  int i = blockIdx.x * 256 + threadIdx.x;
  if (i < total) out[i] = b[i & (OUT_F - 1)];
}

// ------------------------------------------------------------- dense GEMM ---
// Block: 256 threads = 8 waves. Block covers 32 rows; wave w owns N-tile w
// (16 cols) for BOTH 16-row M-tiles -> two independent WMMA accumulator
// chains per wave (hides f32-WMMA latency) and B-fragments shared across the
// two chains (halves B-side LDS reads).
__global__ __launch_bounds__(256) void gcn_gemm(const float* __restrict__ x,
                                                const float* __restrict__ W,
                                                float* __restrict__ support,
                                                int n_nodes) {
  __shared__ float sW[KC][OUT_F];     // 32*128*4 = 16 KB
  __shared__ float sX[MT][KC + 1];    // padded rows to spread LDS banks

  const int m0   = blockIdx.x * MT;
  const int wave = threadIdx.x >> 5;        // 0..7 -> N tile
  const int lane = threadIdx.x & 31;
  const int n0   = wave * 16;

  v8f acc0 = {};
  v8f acc1 = {};

  for (int kc = 0; kc < IN_F; kc += KC) {
    // stage W[kc:kc+KC, :] -> LDS (4096 floats, 16 per thread)
    for (int i = threadIdx.x; i < KC * OUT_F; i += 256) {
      int kr = i >> 7;          // /OUT_F
      int nc = i & (OUT_F - 1);
      sW[kr][nc] = W[(size_t)(kc + kr) * OUT_F + nc];
    }
    // stage x[m0:m0+MT, kc:kc+KC] -> LDS (1024 floats, 4 per thread)
    for (int i = threadIdx.x; i < MT * KC; i += 256) {
      int r = i >> 5;           // /KC
      int c = i & (KC - 1);
      int gr = m0 + r;
      sX[r][c] = (gr < n_nodes) ? x[(size_t)gr * IN_F + kc + c] : 0.0f;
    }
    __syncthreads();

    // 2 x 8 x V_WMMA_F32_16X16X4_F32 over this K-chunk
    const int mrow  = lane & 15;
    const int khalf = (lane >> 4) << 1;   // lanes 0-15: K+0/1, lanes 16-31: K+2/3
#pragma unroll
    for (int kk = 0; kk < KC; kk += 4) {
      const int kb = kk + khalf;
      v2f bfrag;                           // B 4x16: K=kb..kb+1, N=n0+(lane&15)
      bfrag.x = sW[kb][n0 + mrow];
      bfrag.y = sW[kb + 1][n0 + mrow];
      v2f a0;                              // A 16x4, M-tile 0
      a0.x = sX[mrow][kb];
      a0.y = sX[mrow][kb + 1];
      v2f a1;                              // A 16x4, M-tile 1
      a1.x = sX[16 + mrow][kb];
      a1.y = sX[16 + mrow][kb + 1];
      acc0 = __builtin_amdgcn_wmma_f32_16x16x4_f32(
          false, a0, false, bfrag, (short)0, acc0, false, false);
      acc1 = __builtin_amdgcn_wmma_f32_16x16x4_f32(
          false, a1, false, bfrag, (short)0, acc1, false, false);
    }
    __syncthreads();
  }

  // C/D layout: VGPR v -> M = v + (lane>=16 ? 8 : 0), N = lane&15
  const int nn    = lane & 15;
  const int mbase = (lane >> 4) * 8;
#pragma unroll
  for (int v = 0; v < 8; ++v) {
    int m = m0 + mbase + v;
    if (m < n_nodes)      support[(size_t)m * OUT_F + n0 + nn]       = acc0[v];
    if (m + 16 < n_nodes) support[(size_t)(m + 16) * OUT_F + n0 + nn] = acc1[v];
  }
}

// ----------------------------------------------------------------- SpMM -----
// One wave per edge: 32 lanes x float4 = 128 features. support/out are both
// L2-resident (51.2 MB each vs 192 MB L2). Edge index forced wave-uniform via
// readfirstlane so the edge stream goes through scalar (SMEM) loads; prefetch
// the next edge batch (speculative -> silently dropped past the end).
__global__ __launch_bounds__(256) void gcn_spmm(const long long* __restrict__ erow,
                                                const long long* __restrict__ ecol,
                                                const float* __restrict__ eval,
                                                const float* __restrict__ support,
                                                float* __restrict__ out,
                                                int n_edges) {
  const int lane    = threadIdx.x & 31;
  const int wave_id = (blockIdx.x * blockDim.x + threadIdx.x) >> 5;
  const int n_waves = (gridDim.x * blockDim.x) >> 5;

  for (int e = wave_id; e < n_edges; e += n_waves) {
    const int es = __builtin_amdgcn_readfirstlane(e);   // wave-uniform -> s_load
    __builtin_prefetch(erow + es + n_waves, 0, 0);      // global_prefetch_b8
    __builtin_prefetch(ecol + es + n_waves, 0, 0);

    const long long r = erow[es];
    const long long c = ecol[es];
    const float     v = eval[es];

    const float4 s = ((const float4*)(support + (size_t)c * OUT_F))[lane];
    float* dst = out + (size_t)r * OUT_F + lane * 4;
    unsafeAtomicAdd(dst + 0, v * s.x);   // -> global_atomic_add_f32
    unsafeAtomicAdd(dst + 1, v * s.y);
    unsafeAtomicAdd(dst + 2, v * s.z);
    unsafeAtomicAdd(dst + 3, v * s.w);
  }
}

// ------------------------------------------------------------------ launch --
extern "C" void kernel_launch(void* const* d_in, const int* in_sizes, int n_in,
                              void* d_out, int out_size, void* d_ws, size_t ws_size,
                              hipStream_t stream) {
  const float*     x    = (const float*)d_in[0];
  const long long* erow = (const long long*)d_in[1];   // jnp.int64
  const long long* ecol = (const long long*)d_in[2];   // jnp.int64
  const float*     ev   = (const float*)d_in[3];
  const float*     W    = (const float*)d_in[4];
  const float*     b    = (const float*)d_in[5];
  float*           out  = (float*)d_out;

  const int n_nodes = in_sizes[0] / IN_F;
  const int n_edges = in_sizes[1];

  float* support = (float*)d_ws;   // needs n_nodes*OUT_F*4 = 51.2 MB

  // 1) out = bias (also clears harness poison)
  {
    int total = n_nodes * OUT_F;
    gcn_init_out<<<(total + 255) / 256, 256, 0, stream>>>(b, out, total);
  }
  // 2) support = x @ W   (WMMA f32 16x16x4, dual accumulator chains)
  {
    int mtiles = (n_nodes + MT - 1) / MT;    // 3125, exact
    gcn_gemm<<<mtiles, 256, 0, stream>>>(x, W, support, n_nodes);
  }
  // 3) out[row] += val * support[col]
  {
    int blocks = 4096;                       // 32768 waves, ~98 edges each
    gcn_spmm<<<blocks, 256, 0, stream>>>(erow, ecol, ev, support, out, n_edges);
  }
}